// TripletLoss_9354438770967
// MI455X (gfx1250) — compile-verified
//
#include <hip/hip_runtime.h>
#include <hip/hip_bf16.h>
#include <math.h>

// Problem constants (match reference)
#define BN 512          // batch
#define DD 128          // embedding dim
#define MARGIN 0.2f

typedef __attribute__((ext_vector_type(2))) float v2f;
typedef __attribute__((ext_vector_type(8))) float v8f;

// ---------------------------------------------------------------------------
// Zero the two global accumulators (total, count).
// ---------------------------------------------------------------------------
__global__ void init_acc_kernel(float* __restrict__ acc) {
    acc[0] = 0.0f;
    acc[1] = 0.0f;
}

// ---------------------------------------------------------------------------
// Row squared-norms: nrm[i] = sum_k E[i][k]^2   (512 rows x 128)
// ---------------------------------------------------------------------------
__global__ void norms_kernel(const float* __restrict__ E, float* __restrict__ nrm) {
    const int i = blockIdx.x * blockDim.x + threadIdx.x;
    if (i >= BN) return;
    const float4* row = reinterpret_cast<const float4*>(E + i * DD);
    float s = 0.0f;
#pragma unroll
    for (int k = 0; k < DD / 4; ++k) {
        float4 v = row[k];
        s += v.x * v.x + v.y * v.y + v.z * v.z + v.w * v.w;
    }
    nrm[i] = s;
}

// ---------------------------------------------------------------------------
// Gram matrix via V_WMMA_F32_16X16X4_F32, fused distance epilogue.
// One wave computes one 16x16 tile of G = E * E^T.
//   A tile = rows i0..i0+15 of E (16x4 slice per step)
//   B tile = (4x16) = transposed rows j0..j0+15 of E
// Per-lane load pattern (identical for A and B, per ISA 7.12.2 layouts):
//   lane l : row = base + (l & 15), cols k + 2*(l>>4) .. +1   (64-bit load)
// D layout: VGPR v, lane l -> (M = v + 8*(l>>4), N = l & 15)
// Epilogue: sq = nrm_i + nrm_j - 2*dot; dist = sq>0 ? sqrt(sq) : 0; diag = 0.
// ---------------------------------------------------------------------------
__global__ void __launch_bounds__(256)
gram_dist_kernel(const float* __restrict__ E,
                 const float* __restrict__ nrm,
                 float* __restrict__ dist) {
    const int lane = threadIdx.x & 31;
    const int wave = threadIdx.x >> 5;
    const int tile = blockIdx.x * 8 + wave;   // 0..1023 (32x32 tiles)
    const int ti = tile >> 5;
    const int tj = tile & 31;
    const int i0 = ti * 16;
    const int j0 = tj * 16;

    const int r  = lane & 15;         // row within tile
    const int kg = (lane >> 4) * 2;   // K sub-offset: 0 or 2

    const float* __restrict__ arow = E + (i0 + r) * DD + kg;
    const float* __restrict__ brow = E + (j0 + r) * DD + kg;

    v8f c = {0.0f, 0.0f, 0.0f, 0.0f, 0.0f, 0.0f, 0.0f, 0.0f};

#pragma unroll
    for (int k = 0; k < DD; k += 4) {
        v2f a = *reinterpret_cast<const v2f*>(arow + k);
        v2f b = *reinterpret_cast<const v2f*>(brow + k);
        // (neg_a, A, neg_b, B, c_mod, C, reuse_a, reuse_b)
        c = __builtin_amdgcn_wmma_f32_16x16x4_f32(false, a, false, b,
                                                  (short)0, c, false, false);
    }

    const int n  = lane & 15;
    const int mh = (lane >> 4) * 8;
    const float nj = nrm[j0 + n];
#pragma unroll
    for (int v = 0; v < 8; ++v) {
        const int gi = i0 + mh + v;
        const int gj = j0 + n;
        float sq = nrm[gi] + nj - 2.0f * c[v];
        float d  = (sq > 0.0f) ? sqrtf(sq) : 0.0f;
        if (gi == gj) d = 0.0f;   // exact-zero diagonal (matches diff-based ref)
        dist[gi * BN + gj] = d;
    }
}

// ---------------------------------------------------------------------------
// Mining + reduction. One block per anchor row a. Distance row and labels
// staged in LDS. Thread t handles candidate positive p = t.
// Reference semantics replicated exactly (incl. rank-indexing quirk and
// first-negative fallback defaulting to column 0).
// ---------------------------------------------------------------------------
__global__ void __launch_bounds__(BN)
mine_kernel(const float* __restrict__ dist,
            const int* __restrict__ labels,
            float* __restrict__ acc) {
    __shared__ float drow[BN];
    __shared__ int   lab[BN];
    __shared__ float s_sum;
    __shared__ float s_cnt;
    __shared__ int   s_j0;

    const int a = blockIdx.x;
    const int t = threadIdx.x;

    drow[t] = dist[a * BN + t];
    lab[t]  = labels[t];
    if (t == 0) { s_sum = 0.0f; s_cnt = 0.0f; }
    __syncthreads();

    const int la = lab[a];

    if (t == 0) {
        // j0 = argmax(neg) = first negative column (0 if none exist)
        int j0 = 0;
        for (int j = 0; j < BN; ++j) {
            if (lab[j] != la) { j0 = j; break; }
        }
        s_j0 = j0;
    }
    __syncthreads();

    const int p = t;
    if (lab[p] == la && a < p) {          // pair_valid
        const float dap = drow[p];
        const float hi  = dap + MARGIN;

        int jmin = -1;
        for (int j = 0; j < BN; ++j) {
            if (lab[j] != la) {
                const float dj = drow[j];
                if (dj > dap && dj < hi) { jmin = j; break; }
            }
        }

        float dan;
        if (jmin >= 0) {
            // rank of jmin among negatives in [0, jmin), then re-index row
            int rank = 0;
            for (int j = 0; j < jmin; ++j) rank += (lab[j] != la) ? 1 : 0;
            dan = drow[rank];
        } else {
            dan = drow[s_j0];
        }

        float v = dap * dap - dan * dan + MARGIN;
        if (v > 0.0f) atomicAdd(&s_sum, v);   // relu contribution
        atomicAdd(&s_cnt, 1.0f);              // count valid pair regardless
    }
    __syncthreads();

    if (t == 0 && (s_sum != 0.0f || s_cnt != 0.0f)) {
        atomicAdd(&acc[0], s_sum);
        atomicAdd(&acc[1], s_cnt);
    }
}

// ---------------------------------------------------------------------------
// Finalize: scalar loss = total / count
// ---------------------------------------------------------------------------
__global__ void finalize_kernel(const float* __restrict__ acc,
                                float* __restrict__ out) {
    out[0] = acc[0] / acc[1];
}

// ---------------------------------------------------------------------------
// Launch
// ---------------------------------------------------------------------------
extern "C" void kernel_launch(void* const* d_in, const int* in_sizes, int n_in,
                              void* d_out, int out_size, void* d_ws, size_t ws_size,
                              hipStream_t stream) {
    const float* E      = (const float*)d_in[0];   // (512, 128) f32
    const int*   labels = (const int*)d_in[1];     // (512,) i32
    float* out = (float*)d_out;                    // scalar f32

    float* ws   = (float*)d_ws;
    float* dist = ws;                  // 512*512 floats (1 MB)
    float* nrm  = ws + BN * BN;        // 512 floats
    float* acc  = nrm + BN;            // 2 floats: total, count

    init_acc_kernel<<<1, 1, 0, stream>>>(acc);
    norms_kernel<<<(BN + 255) / 256, 256, 0, stream>>>(E, nrm);
    gram_dist_kernel<<<(BN / 16) * (BN / 16) / 8, 256, 0, stream>>>(E, nrm, dist);
    mine_kernel<<<BN, BN, 0, stream>>>(dist, labels, acc);
    finalize_kernel<<<1, 1, 0, stream>>>(acc, out);
}